// LIDDetector_20143396618300
// MI455X (gfx1250) — compile-verified
//
#include <hip/hip_runtime.h>
#include <hip/hip_bf16.h>
#include <math.h>

// ---------------------------------------------------------------------------
// LID detector for MI455X (gfx1250, wave32).
// Memory-bound pooling + f32 WMMA (16x16x4) distance GEMM + per-row top-k LID.
// ---------------------------------------------------------------------------

typedef __attribute__((ext_vector_type(2))) float v2f;
typedef __attribute__((ext_vector_type(8))) float v8f;

#define NUM_REF 2000
#define BATCH   256

// ---------------------------------------------------------------------------
// Kernel 1: spatial mean pool. One wave per (b,c) row of HW contiguous floats.
// Lanes issue coalesced float4 loads (HW % 4 == 0 path) or scalar loads.
// ---------------------------------------------------------------------------
__global__ void pool_kernel(const float* __restrict__ feat, float* __restrict__ Q,
                            int HW, float inv_hw)
{
    const int lane = threadIdx.x & 31;
    const int wv   = threadIdx.x >> 5;
    const int row  = blockIdx.x * 8 + wv;           // row in [0, B*C)
    const float* p = feat + (size_t)row * HW;

    float s = 0.0f;
    if ((HW & 3) == 0) {
        const float4* p4 = (const float4*)p;
        const int nv = HW >> 2;
        for (int i = lane; i < nv; i += 32) {
            float4 v = p4[i];
            s += (v.x + v.y) + (v.z + v.w);
        }
    } else {
        for (int i = lane; i < HW; i += 32) s += p[i];
    }
    #pragma unroll
    for (int off = 16; off > 0; off >>= 1) s += __shfl_down(s, off, 32);
    if (lane == 0) Q[row] = s * inv_hw;
}

// ---------------------------------------------------------------------------
// Kernel 2: squared L2 norm per row. One wave per row of C floats.
// ---------------------------------------------------------------------------
__global__ void rownorm_kernel(const float* __restrict__ X, float* __restrict__ out, int C)
{
    const int lane = threadIdx.x & 31;
    const int wv   = threadIdx.x >> 5;
    const int row  = blockIdx.x * 8 + wv;
    const float* p = X + (size_t)row * C;

    float s = 0.0f;
    for (int i = lane; i < C; i += 32) {
        float v = p[i];
        s += v * v;
    }
    #pragma unroll
    for (int off = 16; off > 0; off >>= 1) s += __shfl_down(s, off, 32);
    if (lane == 0) out[row] = s;
}

// ---------------------------------------------------------------------------
// Kernel 3: squared-distance matrix via V_WMMA_F32_16X16X4_F32.
//   D[m][n] = max(||q_m||^2 + ||r_n||^2 - 2 * (q_m . r_n), 0)
// One wave computes one 16x16 tile of Q(256xC) x R^T(Cx2000).
// A layout (32-bit 16x4): lanes 0-15 hold M rows with K=0,1; lanes 16-31 K=2,3.
// B layout mirrors with N columns striped across lanes.
// EXEC is all-ones: grid exactly covers 16 x 125 = 2000 tiles, no divergence.
// ---------------------------------------------------------------------------
__global__ void dist_wmma_kernel(const float* __restrict__ Q, const float* __restrict__ R,
                                 const float* __restrict__ qn, const float* __restrict__ rn,
                                 float* __restrict__ D, int C)
{
    const int lane = threadIdx.x & 31;
    const int wv   = threadIdx.x >> 5;
    const int tile = blockIdx.x * 8 + wv;           // 0..1999
    const int mt   = tile & 15;                      // 16 M-tiles
    const int nt   = tile >> 4;                      // 125 N-tiles
    const int m0   = mt << 4;
    const int n0   = nt << 4;
    const int lm   = lane & 15;
    const int half = lane >> 4;                      // 0: K=0,1   1: K=2,3

    const float* aptr = Q + (size_t)(m0 + lm) * C + half * 2;
    const float* bptr = R + (size_t)(n0 + lm) * C + half * 2;

    v8f c = {0.f, 0.f, 0.f, 0.f, 0.f, 0.f, 0.f, 0.f};
    for (int k = 0; k < C; k += 4) {
        v2f a = *(const v2f*)(aptr + k);             // 8B aligned: C%4==0, k%4==0
        v2f b = *(const v2f*)(bptr + k);
        c = __builtin_amdgcn_wmma_f32_16x16x4_f32(
                /*neg_a=*/false, a, /*neg_b=*/false, b,
                /*c_mod=*/(short)0, c, /*reuse_a=*/false, /*reuse_b=*/false);
    }

    // C/D layout: lanes 0-15 -> M = m0+v, lanes 16-31 -> M = m0+8+v; N = n0+(lane&15)
    const int   n   = n0 + lm;
    const float rnn = rn[n];
    #pragma unroll
    for (int v = 0; v < 8; ++v) {
        const int m  = m0 + v + half * 8;
        float d2 = qn[m] + rnn - 2.0f * c[v];
        D[(size_t)m * NUM_REF + n] = fmaxf(d2, 0.0f);
    }
}

// ---------------------------------------------------------------------------
// Kernel 4: per-row top-(k+1) selection in the d^2 domain + LID estimate.
// One thread per query row (2000 L2-resident floats). sqrt is monotone:
//   sum log(d_i/d_k) = 0.5 * sum log(d2_i/d2_k)  =>  LID = -2k / sum log(d2_i/d2_k)
// k is read from its device pointer (graph-capture safe), capped at 31.
// ---------------------------------------------------------------------------
__global__ void lid_topk_kernel(const float* __restrict__ D, const int* __restrict__ kptr,
                                float* __restrict__ lid, int layer)
{
    const int b = blockIdx.x * blockDim.x + threadIdx.x;
    if (b >= BATCH) return;

    int kk = kptr[0];
    if (kk < 1)  kk = 1;
    if (kk > 31) kk = 31;
    const int len = kk + 1;

    float arr[32];
    for (int i = 0; i < len; ++i) arr[i] = 3.4e38f;

    const float* row = D + (size_t)b * NUM_REF;
    for (int n = 0; n < NUM_REF; ++n) {
        const float v = row[n];
        if (v < arr[len - 1]) {
            int j = len - 1;
            while (j > 0 && arr[j - 1] > v) { arr[j] = arr[j - 1]; --j; }
            arr[j] = v;
        }
    }

    const float d2k = arr[kk];                       // largest of the kept k (drop nearest)
    float s = 0.0f;
    for (int i = 1; i <= kk; ++i) s += logf(arr[i] / d2k);
    lid[b * 4 + layer] = -(2.0f * (float)kk) / s;
}

// ---------------------------------------------------------------------------
// Kernel 5: logistic regression head over the 4 per-layer LID features.
// ---------------------------------------------------------------------------
__global__ void head_kernel(const float* __restrict__ lid, const float* __restrict__ w,
                            const float* __restrict__ bias, float* __restrict__ out)
{
    const int b = threadIdx.x;                       // single block of 256
    float s = bias[0];
    #pragma unroll
    for (int l = 0; l < 4; ++l) s += lid[b * 4 + l] * w[l];
    out[b] = 1.0f / (1.0f + expf(-s));
}

// ---------------------------------------------------------------------------
// Host-side launcher.
// ---------------------------------------------------------------------------
extern "C" void kernel_launch(void* const* d_in, const int* in_sizes, int n_in,
                              void* d_out, int out_size, void* d_ws, size_t ws_size,
                              hipStream_t stream) {
    (void)in_sizes; (void)n_in; (void)out_size; (void)ws_size;

    const float* feat[4] = { (const float*)d_in[0], (const float*)d_in[1],
                             (const float*)d_in[2], (const float*)d_in[3] };
    const float* ref[4]  = { (const float*)d_in[4], (const float*)d_in[5],
                             (const float*)d_in[6], (const float*)d_in[7] };
    const float* reg_w   = (const float*)d_in[8];
    const float* reg_b   = (const float*)d_in[9];
    const int*   kptr    = (const int*)d_in[10];
    float*       out     = (float*)d_out;

    const int Cs[4]  = { 64, 128, 256, 512 };
    const int HWs[4] = { 56 * 56, 28 * 28, 14 * 14, 7 * 7 };

    // Workspace layout (floats): Q[4 layers concat] | qn[4*256] | rn[4*2000]
    //                            | D[4*256*2000]    | lid[256*4]
    float* ws = (float*)d_ws;
    size_t offQ[4];
    size_t acc = 0;
    for (int l = 0; l < 4; ++l) { offQ[l] = acc; acc += (size_t)BATCH * Cs[l]; }
    const size_t offQn  = acc;                       //  +4*256
    const size_t offRn  = offQn + 4 * BATCH;         //  +4*2000
    const size_t offD   = offRn + 4 * NUM_REF;       //  +4*256*2000
    const size_t offLid = offD + 4 * (size_t)BATCH * NUM_REF;

    for (int l = 0; l < 4; ++l) {
        float* Q  = ws + offQ[l];
        float* qn = ws + offQn + (size_t)l * BATCH;
        float* rn = ws + offRn + (size_t)l * NUM_REF;
        float* D  = ws + offD  + (size_t)l * BATCH * NUM_REF;

        // 1) mean pool: one wave per (b,c) row; rows = 256*C (divisible by 8)
        const int rows = BATCH * Cs[l];
        pool_kernel<<<rows / 8, 256, 0, stream>>>(feat[l], Q, HWs[l],
                                                  1.0f / (float)HWs[l]);

        // 2) row norms: q (256 rows) and ref (2000 rows), both divisible by 8
        rownorm_kernel<<<BATCH / 8,   256, 0, stream>>>(Q,      qn, Cs[l]);
        rownorm_kernel<<<NUM_REF / 8, 256, 0, stream>>>(ref[l], rn, Cs[l]);

        // 3) WMMA distance tiles: 16 * 125 = 2000 tiles, 8 waves/block
        dist_wmma_kernel<<<2000 / 8, 256, 0, stream>>>(Q, ref[l], qn, rn, D, Cs[l]);

        // 4) top-(k+1) + LID, one thread per query row
        lid_topk_kernel<<<1, BATCH, 0, stream>>>(D, kptr, ws + offLid, l);
    }

    // 5) logistic head
    head_kernel<<<1, BATCH, 0, stream>>>(ws + offLid, reg_w, reg_b, out);
}